// Net_vanilla_CNN_convert3_3788161155575
// MI455X (gfx1250) — compile-verified
//
#include <hip/hip_runtime.h>
#include <math.h>

typedef __attribute__((ext_vector_type(2))) float v2f;
typedef __attribute__((ext_vector_type(8))) float v8f;

#define Bn 128
#define Hs 28
#define Ws 28
#define Ls 784          // 28*28
#define OSRN 64
#define EM1F 0.36787944117144233f
#define INV_1mEM1 1.5819767068693265f

// pair table for the 45 unique (e,f) products of the 9-vector (padded to 48)
__device__ const int PAIR_E[48] = {0,0,0,0,0,0,0,0,0, 1,1,1,1,1,1,1,1, 2,2,2,2,2,2,2,
                                   3,3,3,3,3,3, 4,4,4,4,4, 5,5,5,5, 6,6,6, 7,7, 8, 0,0,0};
__device__ const int PAIR_F[48] = {0,1,2,3,4,5,6,7,8, 1,2,3,4,5,6,7,8, 2,3,4,5,6,7,8,
                                   3,4,5,6,7,8, 4,5,6,7,8, 5,6,7,8, 6,7,8, 7,8, 8, 0,0,0};

__device__ inline float hash_noise(unsigned idx) {
    unsigned h = idx * 2654435761u;
    h ^= h >> 16; h *= 2246822519u; h ^= h >> 13; h *= 3266489917u; h ^= h >> 16;
    // roughly unit-variance deterministic perturbation
    return ((h & 0xFFFFFFu) * (1.0f / 16777216.0f) - 0.5f) * 3.464101615f;
}

// ---------------- SRN: normalized unfold y[B,9,L] ----------------
__global__ void srn_y_kernel(const float* __restrict__ x, float* __restrict__ y, int total) {
    int i = blockIdx.x * blockDim.x + threadIdx.x;
    if (i >= total) return;
    int b = i / Ls, l = i - b * Ls;
    int oy = l / Ws, ox = l - oy * Ws;
    float v[9];
    float s = 0.f;
    #pragma unroll
    for (int r = 0; r < 9; ++r) {
        int dy = r / 3 - 1, dx = r % 3 - 1;
        int iy = oy + dy, ix = ox + dx;
        float t = 0.f;
        if ((unsigned)iy < (unsigned)Hs && (unsigned)ix < (unsigned)Ws)
            t = x[b * Ls + iy * Ws + ix];
        t += 1e-4f * hash_noise((unsigned)((b * 9 + r) * Ls + l));
        v[r] = t; s += t;
    }
    float mean = s * (1.0f / 9.0f);
    float var = 0.f;
    #pragma unroll
    for (int r = 0; r < 9; ++r) { float d = v[r] - mean; var += d * d; }
    var *= (1.0f / 8.0f);                    // ddof=1
    float inv_g = rsqrtf(var);
    #pragma unroll
    for (int r = 0; r < 9; ++r)
        y[(b * 9 + r) * Ls + l] = v[r] * inv_g;
}

// ---------------- SRN: quadratic-form coefficient matrix C[64,48] ----------------
__global__ void srn_P_kernel(const float* __restrict__ srn_w, float* __restrict__ Cmat) {
    int o = blockIdx.x * blockDim.x + threadIdx.x;
    if (o >= OSRN) return;
    float X[9][4];
    for (int i = 0; i < 9; ++i) {
        X[i][0] = 1.0f;
        for (int s = 0; s < 3; ++s) X[i][1 + s] = srn_w[o * 27 + i * 3 + s];
    }
    // cov = X^T X  (4x4)
    float aug[4][8];
    for (int a = 0; a < 4; ++a)
        for (int c = 0; c < 4; ++c) {
            float acc = 0.f;
            for (int i = 0; i < 9; ++i) acc += X[i][a] * X[i][c];
            aug[a][c] = acc;
            aug[a][4 + c] = (a == c) ? 1.0f : 0.0f;
        }
    // Gauss-Jordan inverse (SPD, no pivoting)
    for (int p = 0; p < 4; ++p) {
        float piv = 1.0f / aug[p][p];
        for (int c = 0; c < 8; ++c) aug[p][c] *= piv;
        for (int r = 0; r < 4; ++r) {
            if (r == p) continue;
            float f = aug[r][p];
            for (int c = 0; c < 8; ++c) aug[r][c] -= f * aug[p][c];
        }
    }
    // P = I - X inv X^T ; coeffs for quadratic form, off-diagonals doubled, /9
    for (int p = 0; p < 48; ++p) {
        int e = PAIR_E[p], f = PAIR_F[p];
        float h = 0.f;
        for (int a = 0; a < 4; ++a)
            for (int c = 0; c < 4; ++c)
                h += X[e][a] * aug[a][4 + c] * X[f][c];
        float Pef = ((e == f) ? 1.0f : 0.0f) - h;
        float coef = Pef * ((e == f) ? 1.0f : 2.0f) * (1.0f / 9.0f);
        Cmat[o * 48 + p] = (p < 45) ? coef : 0.0f;
    }
}

// ---------------- SRN err GEMM via WMMA f32 16x16x4 (M=64, N=B*L, K=48) ----------------
__global__ void srn_gemm_wmma_kernel(const float* __restrict__ y,
                                     const float* __restrict__ Cmat,
                                     float* __restrict__ h0) {
    const int N = Bn * Ls;
    const int mtiles = OSRN / 16;           // 4
    int wave = blockIdx.x * (blockDim.x >> 5) + (threadIdx.x >> 5);
    if (wave >= mtiles * (N / 16)) return;
    int mt = wave % mtiles, nt = wave / mtiles;
    int lane = threadIdx.x & 31;
    int half = lane >> 4, lr = lane & 15;
    int n = nt * 16 + lr;
    int b = n / Ls, l = n - b * Ls;
    const float* yb = y + b * 9 * Ls + l;

    v8f acc = {};
    #pragma unroll
    for (int k = 0; k < 48; k += 4) {
        int ka = k + 2 * half;
        v2f a, bf;
        a.x = Cmat[(mt * 16 + lr) * 48 + ka];
        a.y = Cmat[(mt * 16 + lr) * 48 + ka + 1];
        int e0 = PAIR_E[ka],     f0 = PAIR_F[ka];
        int e1 = PAIR_E[ka + 1], f1 = PAIR_F[ka + 1];
        bf.x = yb[e0 * Ls] * yb[f0 * Ls];
        bf.y = yb[e1 * Ls] * yb[f1 * Ls];
        acc = __builtin_amdgcn_wmma_f32_16x16x4_f32(false, a, false, bf,
                                                    (short)0, acc, false, false);
    }
    #pragma unroll
    for (int j = 0; j < 8; ++j) {
        int m = mt * 16 + half * 8 + j;
        float err = acc[j];
        float A = (__expf(-err) - EM1F) * INV_1mEM1 - 0.5f;
        h0[(b * OSRN + m) * Ls + l] = A;
    }
}

// ---------------- implicit-GEMM 3x3 conv via WMMA f32 16x16x4 ----------------
__device__ inline float im2col_val(const float* __restrict__ in, int bidx, int CIN,
                                   int H, int W, int kin, int oy, int ox) {
    int ic = kin / 9, r = kin - ic * 9;
    int dy = r / 3 - 1, dx = r % 3 - 1;
    int iy = oy + dy, ix = ox + dx;
    if ((unsigned)iy < (unsigned)H && (unsigned)ix < (unsigned)W)
        return in[((bidx * CIN + ic) * H + iy) * W + ix];
    return 0.0f;
}

__global__ void conv3x3_wmma_kernel(const float* __restrict__ in,
                                    const float* __restrict__ w,
                                    const float* __restrict__ bias,
                                    float* __restrict__ out,
                                    int CIN, int OC, int H, int W) {
    int HW = H * W;
    int N = Bn * HW;                         // multiple of 16 for all layers
    int mtiles = OC >> 4;
    int wave = blockIdx.x * (blockDim.x >> 5) + (threadIdx.x >> 5);
    if (wave >= mtiles * (N >> 4)) return;
    int mt = wave % mtiles, nt = wave / mtiles;
    int lane = threadIdx.x & 31;
    int half = lane >> 4, lr = lane & 15;
    int n = nt * 16 + lr;
    int bidx = n / HW, pix = n - bidx * HW;
    int oy = pix / W, ox = pix - oy * W;
    int Kdim = CIN * 9;
    int row = mt * 16 + lr;

    v8f acc = {};
    for (int k = 0; k < Kdim; k += 4) {
        int ka = k + 2 * half;
        v2f a, bf;
        a.x = w[row * Kdim + ka];
        a.y = w[row * Kdim + ka + 1];
        bf.x = im2col_val(in, bidx, CIN, H, W, ka,     oy, ox);
        bf.y = im2col_val(in, bidx, CIN, H, W, ka + 1, oy, ox);
        acc = __builtin_amdgcn_wmma_f32_16x16x4_f32(false, a, false, bf,
                                                    (short)0, acc, false, false);
    }
    #pragma unroll
    for (int j = 0; j < 8; ++j) {
        int m = mt * 16 + half * 8 + j;
        out[(bidx * OC + m) * HW + pix] = acc[j] + bias[m];
    }
}

// ---------------- BatchNorm: per-channel batch stats (one block per channel) ----------------
__global__ void bn_stats_kernel(const float* __restrict__ x, float* __restrict__ stats,
                                int C, int HW) {
    __shared__ float sh[256], sh2[256];
    int c = blockIdx.x;
    int n = Bn * HW;
    float s = 0.f, s2 = 0.f;
    for (int i = threadIdx.x; i < n; i += blockDim.x) {
        int b = i / HW, p = i - b * HW;
        float v = x[(b * C + c) * HW + p];
        s += v; s2 += v * v;
    }
    sh[threadIdx.x] = s; sh2[threadIdx.x] = s2;
    __syncthreads();
    for (int o = blockDim.x >> 1; o > 0; o >>= 1) {
        if (threadIdx.x < (unsigned)o) {
            sh[threadIdx.x]  += sh[threadIdx.x + o];
            sh2[threadIdx.x] += sh2[threadIdx.x + o];
        }
        __syncthreads();
    }
    if (threadIdx.x == 0) {
        float m = sh[0] / (float)n;
        stats[2 * c]     = m;
        stats[2 * c + 1] = sh2[0] / (float)n - m * m;   // biased variance
    }
}

__global__ void bn_apply_kernel(float* __restrict__ x, const float* __restrict__ stats,
                                const float* __restrict__ g, const float* __restrict__ b,
                                int total, int C, int HW, int relu) {
    int i = blockIdx.x * blockDim.x + threadIdx.x;
    if (i >= total) return;
    int c = (i / HW) % C;
    float m = stats[2 * c], v = stats[2 * c + 1];
    float val = (x[i] - m) * rsqrtf(v + 1e-5f) * g[c] + b[c];
    if (relu) val = fmaxf(val, 0.0f);
    x[i] = val;
}

// ---------------- pooling ----------------
__global__ void avgpool2_kernel(const float* __restrict__ in, float* __restrict__ out,
                                int BC, int H, int W) {
    int OH = H >> 1, OW = W >> 1;
    int total = BC * OH * OW;
    int i = blockIdx.x * blockDim.x + threadIdx.x;
    if (i >= total) return;
    int bc = i / (OH * OW), p = i - bc * (OH * OW);
    int oy = p / OW, ox = p - oy * OW;
    const float* s = in + (size_t)bc * H * W + (2 * oy) * W + 2 * ox;
    out[i] = 0.25f * (s[0] + s[1] + s[W] + s[W + 1]);
}

__global__ void maxpool2p1_kernel(const float* __restrict__ in, float* __restrict__ out,
                                  int BC, int H, int W) {
    int OH = (H + 2 - 2) / 2 + 1, OW = (W + 2 - 2) / 2 + 1;  // k=2,s=2,pad=1
    int total = BC * OH * OW;
    int i = blockIdx.x * blockDim.x + threadIdx.x;
    if (i >= total) return;
    int bc = i / (OH * OW), p = i - bc * (OH * OW);
    int oy = p / OW, ox = p - oy * OW;
    int iy0 = oy * 2 - 1, ix0 = ox * 2 - 1;
    const float* s = in + (size_t)bc * H * W;
    float m = -3.0e38f;
    for (int dy = 0; dy < 2; ++dy)
        for (int dx = 0; dx < 2; ++dx) {
            int iy = iy0 + dy, ix = ix0 + dx;
            if ((unsigned)iy < (unsigned)H && (unsigned)ix < (unsigned)W)
                m = fmaxf(m, s[iy * W + ix]);
        }
    out[i] = m;
}

__global__ void maxpool4_kernel(const float* __restrict__ in, float* __restrict__ out1,
                                float* __restrict__ out2, int BC) {
    int i = blockIdx.x * blockDim.x + threadIdx.x;
    if (i >= BC) return;
    const float* s = in + (size_t)i * 16;
    float m = s[0];
    #pragma unroll
    for (int j = 1; j < 16; ++j) m = fmaxf(m, s[j]);
    out1[i] = m;
    out2[i] = m;
}

// ---------------- small FC (1x1 conv) ----------------
__global__ void fc_kernel(const float* __restrict__ in, const float* __restrict__ w,
                          const float* __restrict__ bias, float* __restrict__ out,
                          int IC, int OC) {
    int i = blockIdx.x * blockDim.x + threadIdx.x;
    if (i >= Bn * OC) return;
    int b = i / OC, oc = i - b * OC;
    float s = bias[oc];
    const float* wr = w + (size_t)oc * IC;
    const float* xr = in + (size_t)b * IC;
    for (int k = 0; k < IC; ++k) s += wr[k] * xr[k];
    out[b * OC + oc] = s;
}

// ---------------- host launch ----------------
extern "C" void kernel_launch(void* const* d_in, const int* in_sizes, int n_in,
                              void* d_out, int out_size, void* d_ws, size_t ws_size,
                              hipStream_t stream) {
    const float* x       = (const float*)d_in[0];
    const float* srn_w   = (const float*)d_in[1];
    const float* conv2_w = (const float*)d_in[2];
    const float* conv2_b = (const float*)d_in[3];
    const float* conv3_w = (const float*)d_in[4];
    const float* conv3_b = (const float*)d_in[5];
    const float* conv4_w = (const float*)d_in[6];
    const float* conv4_b = (const float*)d_in[7];
    const float* fc1_w   = (const float*)d_in[8];
    const float* fc1_b   = (const float*)d_in[9];
    const float* fc2_w   = (const float*)d_in[10];
    const float* fc2_b   = (const float*)d_in[11];
    const float* bn1_g = (const float*)d_in[12]; const float* bn1_b = (const float*)d_in[13];
    const float* bn2_g = (const float*)d_in[14]; const float* bn2_b = (const float*)d_in[15];
    const float* bn3_g = (const float*)d_in[16]; const float* bn3_b = (const float*)d_in[17];
    const float* bn4_g = (const float*)d_in[18]; const float* bn4_b = (const float*)d_in[19];
    const float* bnfc_g = (const float*)d_in[20]; const float* bnfc_b = (const float*)d_in[21];
    float* out = (float*)d_out;

    float* ws = (float*)d_ws;
    // workspace layout (floats)
    float* w_y    = ws;                          // 903168   [B,9,784]
    float* w_C    = w_y    + 903168;             // 3072     [64,48]
    float* w_h0   = w_C    + 3072;               // 6422528  [B,64,28,28]
    float* w_h1   = w_h0   + 6422528;            // 1605632  [B,64,14,14]
    float* w_c2   = w_h1   + 1605632;            // 3211264  [B,128,14,14]
    float* w_h2   = w_c2   + 3211264;            // 802816   [B,128,7,7]
    float* w_c3   = w_h2   + 802816;             // 802816   [B,128,7,7]
    float* w_h3   = w_c3   + 802816;             // 262144   [B,128,4,4]
    float* w_c4   = w_h3   + 262144;             // 262144   [B,128,4,4]
    float* w_h4   = w_c4   + 262144;             // 16384    [B,128]
    float* w_fc1  = w_h4   + 16384;              // 16384    [B,128]
    float* w_st   = w_fc1  + 16384;              // 256      per-channel mean/var

    auto cdiv = [](int a, int b) { return (a + b - 1) / b; };
    const int T = 256;

    // --- SRN ---
    srn_y_kernel<<<cdiv(Bn * Ls, T), T, 0, stream>>>(x, w_y, Bn * Ls);
    srn_P_kernel<<<1, 64, 0, stream>>>(srn_w, w_C);
    {
        int waves = (OSRN / 16) * ((Bn * Ls) / 16);           // 25088
        srn_gemm_wmma_kernel<<<cdiv(waves * 32, 128), 128, 0, stream>>>(w_y, w_C, w_h0);
    }
    // avgpool 28->14, bn1 (no relu)
    avgpool2_kernel<<<cdiv(Bn * 64 * 196, T), T, 0, stream>>>(w_h0, w_h1, Bn * 64, 28, 28);
    bn_stats_kernel<<<64, T, 0, stream>>>(w_h1, w_st, 64, 196);
    bn_apply_kernel<<<cdiv(Bn * 64 * 196, T), T, 0, stream>>>(w_h1, w_st, bn1_g, bn1_b,
                                                              Bn * 64 * 196, 64, 196, 0);
    // --- conv2: 64->128 @14x14, bn+relu, avgpool ->7x7 ---
    {
        int waves = (128 / 16) * ((Bn * 196) / 16);           // 12544
        conv3x3_wmma_kernel<<<cdiv(waves * 32, 128), 128, 0, stream>>>(
            w_h1, conv2_w, conv2_b, w_c2, 64, 128, 14, 14);
    }
    bn_stats_kernel<<<128, T, 0, stream>>>(w_c2, w_st, 128, 196);
    bn_apply_kernel<<<cdiv(Bn * 128 * 196, T), T, 0, stream>>>(w_c2, w_st, bn2_g, bn2_b,
                                                               Bn * 128 * 196, 128, 196, 1);
    avgpool2_kernel<<<cdiv(Bn * 128 * 49, T), T, 0, stream>>>(w_c2, w_h2, Bn * 128, 14, 14);
    // --- conv3: 128->128 @7x7, bn+relu, maxpool(2,pad1) ->4x4 ---
    {
        int waves = (128 / 16) * ((Bn * 49) / 16);            // 3136
        conv3x3_wmma_kernel<<<cdiv(waves * 32, 128), 128, 0, stream>>>(
            w_h2, conv3_w, conv3_b, w_c3, 128, 128, 7, 7);
    }
    bn_stats_kernel<<<128, T, 0, stream>>>(w_c3, w_st, 128, 49);
    bn_apply_kernel<<<cdiv(Bn * 128 * 49, T), T, 0, stream>>>(w_c3, w_st, bn3_g, bn3_b,
                                                              Bn * 128 * 49, 128, 49, 1);
    maxpool2p1_kernel<<<cdiv(Bn * 128 * 16, T), T, 0, stream>>>(w_c3, w_h3, Bn * 128, 7, 7);
    // --- conv4: 128->128 @4x4, bn+relu, maxpool(4) ->1x1 ---
    {
        int waves = (128 / 16) * ((Bn * 16) / 16);            // 1024
        conv3x3_wmma_kernel<<<cdiv(waves * 32, 128), 128, 0, stream>>>(
            w_h3, conv4_w, conv4_b, w_c4, 128, 128, 4, 4);
    }
    bn_stats_kernel<<<128, T, 0, stream>>>(w_c4, w_st, 128, 16);
    bn_apply_kernel<<<cdiv(Bn * 128 * 16, T), T, 0, stream>>>(w_c4, w_st, bn4_g, bn4_b,
                                                              Bn * 128 * 16, 128, 16, 1);
    maxpool4_kernel<<<cdiv(Bn * 128, T), T, 0, stream>>>(w_c4, w_h4, out + Bn * 10, Bn * 128);
    // --- fc1 (1x1 conv) + bn + relu ---
    fc_kernel<<<cdiv(Bn * 128, T), T, 0, stream>>>(w_h4, fc1_w, fc1_b, w_fc1, 128, 128);
    bn_stats_kernel<<<128, T, 0, stream>>>(w_fc1, w_st, 128, 1);
    bn_apply_kernel<<<cdiv(Bn * 128, T), T, 0, stream>>>(w_fc1, w_st, bnfc_g, bnfc_b,
                                                         Bn * 128, 128, 1, 1);
    // --- fc2 -> logits [B,10] ---
    fc_kernel<<<cdiv(Bn * 10, T), T, 0, stream>>>(w_fc1, fc2_w, fc2_b, out, 128, 10);
}